// GraphSAGE_50792283242722
// MI455X (gfx1250) — compile-verified
//
#include <hip/hip_runtime.h>
#include <hip/hip_bf16.h>
#include <math.h>

typedef __attribute__((ext_vector_type(2))) float v2f;
typedef __attribute__((ext_vector_type(8))) float v8f;

#define C 64  // channels (IN==HID==OUT==64)

// ---------------- softmax prep kernels ----------------

// m = 1.0 (self-loop weight is the guaranteed member), s = 0, cnt = 1 (self loop)
__global__ void init_node_kernel(float* __restrict__ m, float* __restrict__ s,
                                 float* __restrict__ cnt, int n) {
  int i = blockIdx.x * blockDim.x + threadIdx.x;
  if (i >= n) return;
  m[i] = 1.0f;
  s[i] = 0.0f;
  cnt[i] = 1.0f;
}

__global__ void zero_kernel(float* __restrict__ p, long long n) {
  long long i = (long long)blockIdx.x * blockDim.x + threadIdx.x;
  if (i < n) p[i] = 0.0f;
}

// segment max by dst (non-negative floats: int-compare == float-compare)
__global__ void edge_max_kernel(const long long* __restrict__ dst,
                                const float* __restrict__ w,
                                float* __restrict__ m, int E) {
  int e = blockIdx.x * blockDim.x + threadIdx.x;
  if (e >= E) return;
  atomicMax((int*)&m[(int)dst[e]], __float_as_int(w[e]));
}

// alpha_raw = exp(w - m[dst]); s[dst] += alpha_raw; cnt[dst] += 1
__global__ void edge_expsum_kernel(const long long* __restrict__ dst,
                                   const float* __restrict__ w,
                                   const float* __restrict__ m,
                                   float* __restrict__ alpha,
                                   float* __restrict__ s,
                                   float* __restrict__ cnt, int E) {
  int e = blockIdx.x * blockDim.x + threadIdx.x;
  if (e >= E) return;
  int d = (int)dst[e];
  float ew = expf(w[e] - m[d]);
  alpha[e] = ew;
  atomicAdd(&s[d], ew);
  atomicAdd(&cnt[d], 1.0f);
}

// self-loop term (runs after edge_expsum, same stream -> plain add is safe)
__global__ void self_expsum_kernel(const float* __restrict__ m,
                                   float* __restrict__ aself,
                                   float* __restrict__ s, int n) {
  int i = blockIdx.x * blockDim.x + threadIdx.x;
  if (i >= n) return;
  float ew = expf(1.0f - m[i]);
  aself[i] = ew;
  s[i] += ew;
}

// ---------------- edge scatter: agg[dst] += (alpha/s[dst]) * h[src] ----------------
// one wave32 per edge; each lane moves 2 consecutive floats of the 64-ch row.
__global__ void scatter_kernel(const long long* __restrict__ src,
                               const long long* __restrict__ dst,
                               const float* __restrict__ alpha,
                               const float* __restrict__ aself,
                               const float* __restrict__ ssum,
                               const float* __restrict__ h,
                               float* __restrict__ agg, int E, int n) {
  int e = blockIdx.x * (blockDim.x >> 5) + (threadIdx.x >> 5);
  int lane = threadIdx.x & 31;
  if (e >= E + n) return;
  int si, di; float a;
  if (e < E) {
    si = (int)src[e]; di = (int)dst[e]; a = alpha[e];
  } else {
    si = di = e - E; a = aself[si];
  }
  a = a / (ssum[di] + 1e-16f);
  const float2* hs = (const float2*)(h + (size_t)si * C);
  float2 v = hs[lane];
  float* ap = agg + (size_t)di * C + 2 * lane;
  atomicAdd(ap, a * v.x);
  atomicAdd(ap + 1, a * v.y);
}

// ---------------- fp32 WMMA GEMM: D = f(rowscale(A) @ B + bias) ----------------
// A: [n,64], B element (k,o) at B[k*sk + o*so], D: [n,64].
// cnt!=nullptr: scale A row r by 1/max(cnt[r],1). bias!=nullptr: add bias[col].
// relu!=0: clamp at 0. One wave computes a 16x64 strip via V_WMMA_F32_16X16X4_F32.
// B is staged once per block into LDS as float2 pairs (B[2p][col], B[2p+1][col])
// so each fragment read is a single ds_load_b64 with an immediate offset.
__global__ void gemm64_wmma_kernel(const float* __restrict__ A,
                                   const float* __restrict__ B,
                                   const float* __restrict__ bias,
                                   const float* __restrict__ cnt,
                                   float* __restrict__ D,
                                   int n, int sk, int so, int relu) {
  __shared__ float2 Bs[(C / 2) * C];  // [pair p = k/2][col], 16 KB

  // stage B -> LDS (stride-agnostic; one-time, all L2 hits)
  for (int i = threadIdx.x; i < (C / 2) * C; i += blockDim.x) {
    int p = i >> 6;        // k pair
    int col = i & (C - 1); // output column
    float2 v;
    v.x = B[(size_t)(2 * p) * sk + (size_t)col * so];
    v.y = B[(size_t)(2 * p + 1) * sk + (size_t)col * so];
    Bs[i] = v;
  }
  __syncthreads();

  int wave = threadIdx.x >> 5;
  int lane = threadIdx.x & 31;
  int rt = blockIdx.x * (blockDim.x >> 5) + wave;   // row-tile index
  int r0 = rt * 16;
  if (r0 + 16 > n) return;                          // whole wave drops together
  int half = lane >> 4;                             // 0: K={0,1}, 1: K={2,3}
  int lm = lane & 15;

  int arow = r0 + lm;
  float scale = 1.0f;
  if (cnt) scale = 1.0f / fmaxf(cnt[arow], 1.0f);
  const float* Ap = A + (size_t)arow * C;
  int bbase = half * 64 + lm;                       // lane-constant LDS index

  v8f acc0 = {}, acc1 = {}, acc2 = {}, acc3 = {};
#pragma unroll
  for (int k0 = 0; k0 < C; k0 += 4) {
    int ka = k0 + half * 2;
    v2f a;
    a.x = Ap[ka] * scale;
    a.y = Ap[ka + 1] * scale;
    // four 16-col tiles of B from LDS: index = k0*32 + bbase + t*16 (imm offsets)
    float2 p0 = Bs[k0 * 32 + bbase + 0];
    float2 p1 = Bs[k0 * 32 + bbase + 16];
    float2 p2 = Bs[k0 * 32 + bbase + 32];
    float2 p3 = Bs[k0 * 32 + bbase + 48];
    v2f b0; b0.x = p0.x; b0.y = p0.y;
    v2f b1; b1.x = p1.x; b1.y = p1.y;
    v2f b2; b2.x = p2.x; b2.y = p2.y;
    v2f b3; b3.x = p3.x; b3.y = p3.y;
    acc0 = __builtin_amdgcn_wmma_f32_16x16x4_f32(false, a, false, b0, (short)0, acc0, false, false);
    acc1 = __builtin_amdgcn_wmma_f32_16x16x4_f32(false, a, false, b1, (short)0, acc1, false, false);
    acc2 = __builtin_amdgcn_wmma_f32_16x16x4_f32(false, a, false, b2, (short)0, acc2, false, false);
    acc3 = __builtin_amdgcn_wmma_f32_16x16x4_f32(false, a, false, b3, (short)0, acc3, false, false);
  }

  // C/D layout: vgpr v, lane: row = r0 + v + 8*half, col = tile*16 + lm
#pragma unroll
  for (int v = 0; v < 8; v++) {
    int row = r0 + v + 8 * half;
    float* Drow = D + (size_t)row * C;
#pragma unroll
    for (int t = 0; t < 4; t++) {
      int col = t * 16 + lm;
      float val = (t == 0 ? acc0[v] : t == 1 ? acc1[v] : t == 2 ? acc2[v] : acc3[v]);
      if (bias) val += bias[col];
      if (relu) val = fmaxf(val, 0.0f);
      Drow[col] = val;
    }
  }
}

// ---------------- host-side orchestration ----------------

extern "C" void kernel_launch(void* const* d_in, const int* in_sizes, int n_in,
                              void* d_out, int out_size, void* d_ws, size_t ws_size,
                              hipStream_t stream) {
  const float*     x   = (const float*)d_in[0];
  const long long* ei  = (const long long*)d_in[1];   // int64 [2,E]
  const float*     ew  = (const float*)d_in[2];
  const float*     l1w = (const float*)d_in[3];
  const float*     l1b = (const float*)d_in[4];
  const float*     W1  = (const float*)d_in[5];
  const float*     l2w = (const float*)d_in[6];
  const float*     l2b = (const float*)d_in[7];
  const float*     W2  = (const float*)d_in[8];
  float* out = (float*)d_out;

  const int n = in_sizes[0] / C;      // 100000
  const int E = in_sizes[2];          // 1600000
  const long long* src = ei;
  const long long* dst = ei + E;

  // workspace layout (floats)
  float* m     = (float*)d_ws;
  float* ssum  = m + n;
  float* cnt   = ssum + n;
  float* alpha = cnt + n;
  float* aself = alpha + E;
  float* bufA  = aself + n;                 // h = lin(x)
  float* bufB  = bufA + (size_t)n * C;      // agg
  float* bufC  = bufB + (size_t)n * C;      // relu(layer1 out)

  const int T = 256;
  int gbN  = (n + T - 1) / T;
  int gbE  = (E + T - 1) / T;
  long long feat = (long long)n * C;
  int gbF  = (int)((feat + T - 1) / T);
  int tiles = n / 16;                       // full 16-row tiles (n % 16 == 0 here)
  int gbG  = (tiles + 7) / 8;               // 8 waves / 256-thread block
  int gbS  = (E + n + 7) / 8;               // 1 wave per edge (+self)

  // shared edge softmax
  init_node_kernel<<<gbN, T, 0, stream>>>(m, ssum, cnt, n);
  edge_max_kernel<<<gbE, T, 0, stream>>>(dst, ew, m, E);
  edge_expsum_kernel<<<gbE, T, 0, stream>>>(dst, ew, m, alpha, ssum, cnt, E);
  self_expsum_kernel<<<gbN, T, 0, stream>>>(m, aself, ssum, n);

  // ---- layer 1 ----
  gemm64_wmma_kernel<<<gbG, T, 0, stream>>>(x, l1w, l1b, nullptr, bufA, n,
                                            /*sk=*/1, /*so=*/C, /*relu=*/0);   // h = x@l1w.T + b
  zero_kernel<<<gbF, T, 0, stream>>>(bufB, feat);
  scatter_kernel<<<gbS, T, 0, stream>>>(src, dst, alpha, aself, ssum, bufA, bufB, E, n);
  gemm64_wmma_kernel<<<gbG, T, 0, stream>>>(bufB, W1, nullptr, cnt, bufC, n,
                                            /*sk=*/C, /*so=*/1, /*relu=*/1);   // relu((agg/cnt)@W1)

  // ---- layer 2 ----
  gemm64_wmma_kernel<<<gbG, T, 0, stream>>>(bufC, l2w, l2b, nullptr, bufA, n,
                                            /*sk=*/1, /*so=*/C, /*relu=*/0);   // h = relu_out@l2w.T + b
  zero_kernel<<<gbF, T, 0, stream>>>(bufB, feat);
  scatter_kernel<<<gbS, T, 0, stream>>>(src, dst, alpha, aself, ssum, bufA, bufB, E, n);
  gemm64_wmma_kernel<<<gbG, T, 0, stream>>>(bufB, W2, nullptr, cnt, out, n,
                                            /*sk=*/C, /*so=*/1, /*relu=*/0);   // (agg/cnt)@W2
}